// Entmax_5437428597416
// MI455X (gfx1250) — compile-verified
//
#include <hip/hip_runtime.h>

// Problem constants (match reference)
#define ROWS 2048
#define D 32000
#define N_ITER 100
#define ALPHA_M1 0.25f          // alpha - 1
#define TPB 1024                // 32 wave32 waves per workgroup (one WGP)
#define CHUNKS 8                // 8 x float4 per thread: 1024*8*4 = 32768 slots
#define CHUNK_FLOATS (TPB * 4)  // 4096 floats per chunk
#define VALID_T7 832            // chunk 7 valid for t < 832 (28672 + 832*4 == 32000)
#define NEG_BIG (-1.0e30f)
#define ROW_BYTES (D * 4)       // 128000 B of dynamic LDS for the staged row

typedef float v4f __attribute__((ext_vector_type(4)));
typedef unsigned int u32x4 __attribute__((ext_vector_type(4)));
typedef int i32x4 __attribute__((ext_vector_type(4)));
typedef int i32x8 __attribute__((ext_vector_type(8)));

// Tensor Data Mover path: stage the whole row HBM -> LDS with one TDM
// descriptor issued by a single wave, then pull registers from LDS once.
#if defined(__AMDGCN__) && defined(__has_builtin)
#if __has_builtin(__builtin_amdgcn_tensor_load_to_lds) && \
    __has_builtin(__builtin_amdgcn_s_wait_tensorcnt)
#define USE_TDM 1
#endif
#endif
#ifndef USE_TDM
#define USE_TDM 0
#endif

__device__ __forceinline__ float warpReduceSum(float v) {
#pragma unroll
  for (int off = 16; off > 0; off >>= 1)
    v += __shfl_xor(v, off, 32);
  return v;
}

__device__ __forceinline__ float warpReduceMax(float v) {
#pragma unroll
  for (int off = 16; off > 0; off >>= 1)
    v = fmaxf(v, __shfl_xor(v, off, 32));
  return v;
}

// Two-phase block reduction with broadcast. Exactly 2 barriers; barrier
// ordering makes back-to-back invocations race-free.
__device__ __forceinline__ float blockReduceSumBcast(float v, float* s_part,
                                                     float* s_b, int lane,
                                                     int wid) {
  v = warpReduceSum(v);
  if (lane == 0) s_part[wid] = v;
  __syncthreads();
  if (wid == 0) {
    float w = warpReduceSum(s_part[lane]);
    if (lane == 0) *s_b = w;
  }
  __syncthreads();
  return *s_b;
}

// sum over this thread's 32 elements of relu(X - tau)^4
__device__ __forceinline__ float p4PartialSum(const v4f* X, float tau) {
  float acc = 0.0f;
#pragma unroll
  for (int c = 0; c < CHUNKS; ++c) {
    float z0 = fmaxf(X[c].x - tau, 0.0f);
    float z1 = fmaxf(X[c].y - tau, 0.0f);
    float z2 = fmaxf(X[c].z - tau, 0.0f);
    float z3 = fmaxf(X[c].w - tau, 0.0f);
    float a0 = z0 * z0, a1 = z1 * z1, a2 = z2 * z2, a3 = z3 * z3;
    acc = fmaf(a0, a0, acc);
    acc = fmaf(a1, a1, acc);
    acc = fmaf(a2, a2, acc);
    acc = fmaf(a3, a3, acc);
  }
  return acc;
}

__global__ __launch_bounds__(TPB, 1) void entmax_bisect_kernel(
    const float* __restrict__ in, float* __restrict__ out) {
  __shared__ float s_part[TPB / 32];
  __shared__ float s_b;

  const int row = blockIdx.x;
  const int t = threadIdx.x;
  const int lane = t & 31;
  const int wid = t >> 5;
  const float* rp = in + (size_t)row * D;
  float* op = out + (size_t)row * D;

  v4f X[CHUNKS];

#if USE_TDM
  extern __shared__ __align__(16) float s_row[];

  // ---- One TDM descriptor moves the whole 125 KB row HBM -> LDS.
  // Tensor ops ignore EXEC, so guarding on thread 0 issues it exactly once.
  if (t == 0) {
    const unsigned long long ga = (unsigned long long)(const void*)rp;
    // Generic LDS pointer low 32 bits == LDS byte address (aperture mapping).
    const unsigned lds_addr = (unsigned)(unsigned long long)(const void*)s_row;

    // D# group 0: count=1 (valid), lds_addr, 57-bit global_addr, type=2.
    u32x4 g0;
    g0[0] = 1u;
    g0[1] = lds_addr;
    g0[2] = (unsigned)(ga & 0xFFFFFFFFu);
    g0[3] = (unsigned)((ga >> 32) & 0x1FFFFFFu) | (2u << 30);

    // D# group 1: data_size=4B (2), tensor_dim0 = tile_dim0 = 32000,
    // tensor_dim1 = tile_dim1 = 1, tensor_dim0_stride = 32000.
    i32x8 g1;
    g1[0] = 0x00020000;                 // [17:16] data_size = 2 (4 bytes)
    g1[1] = (int)((unsigned)D << 16);   // [63:48]  tensor_dim0[15:0]
    g1[2] = (int)(1u << 16);            // [95:80]  tensor_dim1 = 1
    g1[3] = (int)((unsigned)D << 16);   // [127:112] tile_dim0 = 32000
    g1[4] = 1;                          // [143:128] tile_dim1 = 1
    g1[5] = D;                          // [191:160] tensor_dim0_stride lo
    g1[6] = 0;
    g1[7] = 0;

    i32x4 z4 = {0, 0, 0, 0};            // groups 2/3 unused (<= 2D tensor)

#if __clang_major__ >= 23
    i32x8 z8 = {0, 0, 0, 0, 0, 0, 0, 0};
    __builtin_amdgcn_tensor_load_to_lds(g0, g1, z4, z4, z8, 0);
#else
    __builtin_amdgcn_tensor_load_to_lds(g0, g1, z4, z4, 0);
#endif
    __builtin_amdgcn_s_wait_tensorcnt(0);
  }
  __syncthreads();  // publish staged LDS row to all 32 waves

  // ---- Pull this thread's 32 elements LDS -> registers (ds_load_b128),
  // pre-scaled by (alpha-1). One-time cost, then LDS is never re-read.
#pragma unroll
  for (int c = 0; c < CHUNKS; ++c) {
    if (c < CHUNKS - 1 || t < VALID_T7) {
      const v4f* p =
          reinterpret_cast<const v4f*>(s_row + c * CHUNK_FLOATS + t * 4);
      X[c] = (*p) * ALPHA_M1;
    } else {
      X[c] = (v4f){NEG_BIG, NEG_BIG, NEG_BIG, NEG_BIG};  // pad: contributes 0
    }
  }
#else
  // ---- Fallback: streaming (NT) 128-bit global loads straight to VGPRs.
#pragma unroll
  for (int c = 0; c < CHUNKS; ++c) {
    if (c < CHUNKS - 1 || t < VALID_T7) {
      const v4f* p =
          reinterpret_cast<const v4f*>(rp + c * CHUNK_FLOATS + t * 4);
      v4f v = __builtin_nontemporal_load(p);
      X[c] = v * ALPHA_M1;
    } else {
      X[c] = (v4f){NEG_BIG, NEG_BIG, NEG_BIG, NEG_BIG};
    }
  }
#endif

  // ---- Row max (over scaled X)
  float m = NEG_BIG;
#pragma unroll
  for (int c = 0; c < CHUNKS; ++c) {
    m = fmaxf(m, fmaxf(fmaxf(X[c].x, X[c].y), fmaxf(X[c].z, X[c].w)));
  }
  m = warpReduceMax(m);
  if (lane == 0) s_part[wid] = m;
  __syncthreads();
  if (wid == 0) {
    float w = warpReduceMax(s_part[lane]);
    if (lane == 0) s_b = w;
  }
  __syncthreads();
  const float max_val = s_b;

  // ---- Bisection (all threads step identically off broadcast values).
  float tau_lo = max_val - 1.0f;
  const float tau_hi = max_val - sqrtf(sqrtf(1.0f / (float)D));
  float dm = tau_hi - tau_lo;
  const float f_lo =
      blockReduceSumBcast(p4PartialSum(X, tau_lo), s_part, &s_b, lane, wid) -
      1.0f;

  float tau_m = tau_lo;
  float S_m = 1.0f;
  for (int it = 0; it < N_ITER; ++it) {
    dm *= 0.5f;
    tau_m = tau_lo + dm;
    S_m = blockReduceSumBcast(p4PartialSum(X, tau_m), s_part, &s_b, lane, wid);
    const float f_m = S_m - 1.0f;
    if (f_m * f_lo >= 0.0f) tau_lo = tau_m;  // faithful to reference
  }

  // ---- Output: p_m (from the LAST iteration's tau_m) / sum(p_m).
  const float inv = 1.0f / S_m;
#pragma unroll
  for (int c = 0; c < CHUNKS; ++c) {
    if (c < CHUNKS - 1 || t < VALID_T7) {
      float z0 = fmaxf(X[c].x - tau_m, 0.0f);
      float z1 = fmaxf(X[c].y - tau_m, 0.0f);
      float z2 = fmaxf(X[c].z - tau_m, 0.0f);
      float z3 = fmaxf(X[c].w - tau_m, 0.0f);
      float a0 = z0 * z0, a1 = z1 * z1, a2 = z2 * z2, a3 = z3 * z3;
      v4f o = {a0 * a0 * inv, a1 * a1 * inv, a2 * a2 * inv, a3 * a3 * inv};
      v4f* p = reinterpret_cast<v4f*>(op + c * CHUNK_FLOATS + t * 4);
      __builtin_nontemporal_store(o, p);
    }
  }
}

extern "C" void kernel_launch(void* const* d_in, const int* in_sizes, int n_in,
                              void* d_out, int out_size, void* d_ws,
                              size_t ws_size, hipStream_t stream) {
  (void)in_sizes;
  (void)n_in;
  (void)out_size;
  (void)d_ws;
  (void)ws_size;
  const float* x = (const float*)d_in[0];
  float* out = (float*)d_out;
  entmax_bisect_kernel<<<dim3(ROWS), dim3(TPB), ROW_BYTES, stream>>>(x, out);
}